// Generator_29016799052023
// MI455X (gfx1250) — compile-verified
//
#include <hip/hip_runtime.h>
#include <hip/hip_bf16.h>
#include <math.h>

// ---------------------------------------------------------------------------
// Graph generator for MI455X (gfx1250, wave32, WMMA).
//
// Sizes: B=32, LATENT=128, H_NODE=128, MAX_NODES=512, NODE_DIM=8, H_EDGE=64.
// d_out = [nodes (32*512*8 f32)] ++ [adj (32*512*512 f32)]
//
// Pipeline:
//   K1: h = relu(z @ W1 + b1)                      (WMMA f32 16x16x4, M32 K128 N128)
//   K2: nodes = h @ W2 + b2        -> d_out        (WMMA f32 16x16x4, M32 K128 N4096)
//   K3: An = nodes @ E1[0:8] + eb1, Bn = nodes @ E1[8:16]
//                                                 (WMMA f32 16x16x4, M16384 K8 N64)
//   K4: for i<j: p = sigmoid(sum_k relu(An[i,k]+Bn[j,k])*E2[k] + eb2)
//       adj[i,j]=adj[j,i]=p, diag=0              (VALU + LDS tiles; not WMMA-able)
// ---------------------------------------------------------------------------

#if defined(__HIP_DEVICE_COMPILE__)
#if !__has_builtin(__builtin_amdgcn_wmma_f32_16x16x4_f32)
#error "missing __builtin_amdgcn_wmma_f32_16x16x4_f32 on this toolchain (device)"
#endif
#endif

typedef __attribute__((ext_vector_type(2))) float v2f;
typedef __attribute__((ext_vector_type(8))) float v8f;

#define BATCH     32
#define LATENT    128
#define H_NODE    128
#define MAX_NODES 512
#define NODE_DIM  8
#define H_EDGE    64
#define NODE_OUT  (MAX_NODES * NODE_DIM)   // 4096

__device__ __forceinline__ v8f wmma_f32_k4(v2f a, v2f b, v8f c) {
#if defined(__HIP_DEVICE_COMPILE__)
  // D = A(16x4,f32) * B(4x16,f32) + C(16x16,f32)
  return __builtin_amdgcn_wmma_f32_16x16x4_f32(
      /*neg_a=*/false, a, /*neg_b=*/false, b,
      /*c_mod=*/(short)0, c, /*reuse_a=*/false, /*reuse_b=*/false);
#else
  return c;
#endif
}

// --------------------------- K1: h = relu(z@W1+b1) --------------------------
// M=32, K=128, N=128. 2x8 = 16 wave tiles -> one block of 512 threads.
__global__ void __launch_bounds__(512)
k1_node_hidden(const float* __restrict__ z, const float* __restrict__ W1,
               const float* __restrict__ b1, float* __restrict__ h) {
  const int wave = threadIdx.x >> 5;
  const int lane = threadIdx.x & 31;
  const int mt = wave >> 3;      // 0..1
  const int nt = wave & 7;       // 0..7
  const int lrow = lane & 15;
  const int khalf = (lane >> 4) << 1;          // 0 or 2
  const int arow = mt * 16 + lrow;             // A row (z row)
  const int bcol = nt * 16 + lrow;             // B col (W1 col)

  v8f acc = {};
  for (int k0 = 0; k0 < LATENT; k0 += 4) {
    v2f a, b;
    a.x = z[arow * LATENT + k0 + khalf + 0];
    a.y = z[arow * LATENT + k0 + khalf + 1];
    b.x = W1[(k0 + khalf + 0) * H_NODE + bcol];
    b.y = W1[(k0 + khalf + 1) * H_NODE + bcol];
    acc = wmma_f32_k4(a, b, acc);
  }
  // C/D layout: VGPR r -> M = mt*16 + r (+8 if lane>=16), N = bcol
  const int m_lo = mt * 16 + ((lane >= 16) ? 8 : 0);
  const float bias = b1[bcol];
#pragma unroll
  for (int r = 0; r < 8; ++r) {
    float v = acc[r] + bias;
    h[(m_lo + r) * H_NODE + bcol] = fmaxf(v, 0.0f);
  }
}

// --------------------------- K2: nodes = h@W2+b2 ----------------------------
// M=32, K=128, N=4096. 2x256 = 512 wave tiles -> 64 blocks x 256 threads.
__global__ void __launch_bounds__(256)
k2_node_out(const float* __restrict__ h, const float* __restrict__ W2,
            const float* __restrict__ b2, float* __restrict__ nodes) {
  const int gw = blockIdx.x * 8 + (threadIdx.x >> 5);   // 0..511
  const int lane = threadIdx.x & 31;
  const int mt = gw >> 8;       // 0..1
  const int nt = gw & 255;      // 0..255
  const int lrow = lane & 15;
  const int khalf = (lane >> 4) << 1;
  const int arow = mt * 16 + lrow;
  const int bcol = nt * 16 + lrow;

  v8f acc = {};
  for (int k0 = 0; k0 < H_NODE; k0 += 4) {
    v2f a, b;
    a.x = h[arow * H_NODE + k0 + khalf + 0];
    a.y = h[arow * H_NODE + k0 + khalf + 1];
    b.x = W2[(size_t)(k0 + khalf + 0) * NODE_OUT + bcol];
    b.y = W2[(size_t)(k0 + khalf + 1) * NODE_OUT + bcol];
    acc = wmma_f32_k4(a, b, acc);
  }
  const int m_lo = mt * 16 + ((lane >= 16) ? 8 : 0);
  const float bias = b2[bcol];
#pragma unroll
  for (int r = 0; r < 8; ++r)
    nodes[(size_t)(m_lo + r) * NODE_OUT + bcol] = acc[r] + bias;
}

// ------------------ K3: An = nodes@E1_top + eb1, Bn = nodes@E1_bot ----------
// nodes viewed as [B*512, 8]. M=16384, K=8, N=64. 1024x4 = 4096 wave tiles
// -> 512 blocks x 256 threads. A-fragments shared between the two GEMMs.
__global__ void __launch_bounds__(256)
k3_edge_pre(const float* __restrict__ nodes, const float* __restrict__ E1,
            const float* __restrict__ eb1, float* __restrict__ An,
            float* __restrict__ Bn) {
  const int gw = blockIdx.x * 8 + (threadIdx.x >> 5);   // 0..4095
  const int lane = threadIdx.x & 31;
  const int mt = gw >> 2;       // 0..1023
  const int nt = gw & 3;        // 0..3
  const int lrow = lane & 15;
  const int khalf = (lane >> 4) << 1;
  const int arow = mt * 16 + lrow;
  const int bcol = nt * 16 + lrow;

  v8f accA = {}, accB = {};
  for (int k0 = 0; k0 < NODE_DIM; k0 += 4) {
    v2f a, bt, bb;
    a.x = nodes[(size_t)arow * NODE_DIM + k0 + khalf + 0];
    a.y = nodes[(size_t)arow * NODE_DIM + k0 + khalf + 1];
    bt.x = E1[(k0 + khalf + 0) * H_EDGE + bcol];
    bt.y = E1[(k0 + khalf + 1) * H_EDGE + bcol];
    bb.x = E1[(NODE_DIM + k0 + khalf + 0) * H_EDGE + bcol];
    bb.y = E1[(NODE_DIM + k0 + khalf + 1) * H_EDGE + bcol];
    accA = wmma_f32_k4(a, bt, accA);
    accB = wmma_f32_k4(a, bb, accB);
  }
  const int m_lo = mt * 16 + ((lane >= 16) ? 8 : 0);
  const float bias = eb1[bcol];
#pragma unroll
  for (int r = 0; r < 8; ++r) {
    An[(size_t)(m_lo + r) * H_EDGE + bcol] = accA[r] + bias;
    Bn[(size_t)(m_lo + r) * H_EDGE + bcol] = accB[r];
  }
}

// ------------------------ K4: pairwise edge MLP -> adj ----------------------
// grid = (tj=32, ti=32, b=32); blocks with ti>tj exit (upper-tri tiles only).
// 16x16 pair tile per block, 256 threads, one pair each.
__global__ void __launch_bounds__(256)
k4_edge_adj(const float* __restrict__ An, const float* __restrict__ Bn,
            const float* __restrict__ E2, const float* __restrict__ eb2,
            float* __restrict__ adj) {
  const int tj = blockIdx.x, ti = blockIdx.y, b = blockIdx.z;
  if (ti > tj) return;

  __shared__ float As[16][65];   // stride 65: conflict-free column walks
  __shared__ float Bs[16][65];
  __shared__ float e2s[H_EDGE];
  __shared__ float pt[16][17];   // tile of p for coalesced mirror store

  const int tid = threadIdx.x;
  const float* Ab = An + ((size_t)(b * MAX_NODES + ti * 16)) * H_EDGE;
  const float* Bb = Bn + ((size_t)(b * MAX_NODES + tj * 16)) * H_EDGE;
  for (int idx = tid; idx < 16 * H_EDGE; idx += 256) {
    const int r = idx >> 6, c = idx & 63;
    As[r][c] = Ab[idx];
    Bs[r][c] = Bb[idx];
  }
  if (tid < H_EDGE) e2s[tid] = E2[tid];
  __syncthreads();

  const int i = tid >> 4, j = tid & 15;
  float acc = 0.0f;
#pragma unroll
  for (int k = 0; k < H_EDGE; ++k)
    acc = fmaf(fmaxf(As[i][k] + Bs[j][k], 0.0f), e2s[k], acc);
  const float x = acc + eb2[0];
  const float p = 1.0f / (1.0f + __expf(-x));
  pt[i][j] = p;

  float* adjb = adj + (size_t)b * MAX_NODES * MAX_NODES;
  const int ig = ti * 16 + i, jg = tj * 16 + j;
  if (ti < tj) {
    adjb[(size_t)ig * MAX_NODES + jg] = p;            // upper block, coalesced
  } else {                                            // diagonal tile
    if (i < j)       adjb[(size_t)ig * MAX_NODES + jg] = p;
    else if (i == j) adjb[(size_t)ig * MAX_NODES + jg] = 0.0f;
  }
  __syncthreads();

  // mirror: adj[jg', ig'] via LDS transpose so stores stay coalesced
  const int r = tid >> 4, c = tid & 15;
  const float pv = pt[c][r];
  const int rg = tj * 16 + r, cg = ti * 16 + c;
  if (ti < tj)     adjb[(size_t)rg * MAX_NODES + cg] = pv;
  else if (c < r)  adjb[(size_t)rg * MAX_NODES + cg] = pv;  // diag tile lower
}

// ---------------------------------------------------------------------------
extern "C" void kernel_launch(void* const* d_in, const int* in_sizes, int n_in,
                              void* d_out, int out_size, void* d_ws, size_t ws_size,
                              hipStream_t stream) {
  const float* z   = (const float*)d_in[0];
  const float* W1  = (const float*)d_in[1];
  const float* b1  = (const float*)d_in[2];
  const float* W2  = (const float*)d_in[3];
  const float* b2  = (const float*)d_in[4];
  const float* E1  = (const float*)d_in[5];
  const float* eb1 = (const float*)d_in[6];
  const float* E2  = (const float*)d_in[7];
  const float* eb2 = (const float*)d_in[8];

  float* out_nodes = (float*)d_out;                                // 32*512*8
  float* out_adj   = out_nodes + BATCH * MAX_NODES * NODE_DIM;     // 32*512*512

  float* ws_h  = (float*)d_ws;                                     // 4096 f32
  float* ws_An = ws_h + 4096;                                      // 16384*64
  float* ws_Bn = ws_An + (size_t)BATCH * MAX_NODES * H_EDGE;       // 16384*64

  k1_node_hidden<<<1, 512, 0, stream>>>(z, W1, b1, ws_h);
  k2_node_out<<<64, 256, 0, stream>>>(ws_h, W2, b2, out_nodes);
  k3_edge_pre<<<512, 256, 0, stream>>>(out_nodes, E1, eb1, ws_An, ws_Bn);
  k4_edge_adj<<<dim3(32, 32, 32), 256, 0, stream>>>(ws_An, ws_Bn, E2, eb2, out_adj);
}